// Transform_48533130444816
// MI455X (gfx1250) — compile-verified
//
#include <hip/hip_runtime.h>

typedef __attribute__((ext_vector_type(2))) float v2f;
typedef __attribute__((ext_vector_type(4))) float v4f;
typedef __attribute__((ext_vector_type(8))) float v8f;

#define DIN   128
#define IP    17                       // pad stride for i-dim (16 -> 17)
#define JP    17                       // pad stride for j-dim of M tile
#define REGION_FLOATS (2048 * IP)      // [c = g*128+d][i] : 34816 floats
#define M_OFF  REGION_FLOATS
#define TOTAL_LDS_FLOATS (REGION_FLOATS + 256 * JP)   // + M tile [(i*16+g)][j]

extern __shared__ float lds[];

__global__ __launch_bounds__(512)
void fused_dynconv_kernel(const float* __restrict__ region,
                          const float* __restrict__ convw,
                          const float* __restrict__ convb,
                          float* __restrict__ out)
{
    const int t    = threadIdx.x;      // 0..511, 16 waves of 32
    const int wave = t >> 5;           // 0..15
    const int lane = t & 31;
    const int lo   = lane & 15;        // lane within half
    const int hi   = lane >> 4;        // which 16-lane half
    const long n0  = (long)blockIdx.x * 16;

    // ---- stage 16x16x128 region tile (32768 contiguous floats) into LDS [c][i] ----
    const float* tile = region + n0 * 2048;
    #pragma unroll
    for (int s = 0; s < 16; ++s) {
        int q = s * 512 + t;                 // float4 index
        int f = q * 4;                       // flat float index in tile
        v4f val = *(const v4f*)(tile + f);
        int i = f >> 11;                     // n-local row (f / 2048)
        int c = f & 2047;                    // g*128 + d
        lds[(c + 0) * IP + i] = val.x;
        lds[(c + 1) * IP + i] = val.y;
        lds[(c + 2) * IP + i] = val.z;
        lds[(c + 3) * IP + i] = val.w;
    }
    __syncthreads();

    // ---- phase 1: wave g computes M[i][j] = sum_d A[i][g][d] * W[g][j][d] ----
    const int g = wave;
    v8f acc = {};
    const float* wrow = convw + (g * 16 + lo) * DIN;   // W[g][j=lo][*]
    #pragma unroll 4
    for (int kc = 0; kc < 32; ++kc) {
        int d0 = 4 * kc + hi * 2;            // K chunk: lanes 0-15 -> k0,k1; 16-31 -> k2,k3
        v2f a;
        a.x = lds[(g * DIN + d0    ) * IP + lo];       // A[i=lo][d0]
        a.y = lds[(g * DIN + d0 + 1) * IP + lo];
        v2f b = *(const v2f*)(wrow + d0);              // B[k][j=lo] = W[g][lo][d0..d0+1]
        acc = __builtin_amdgcn_wmma_f32_16x16x4_f32(false, a, false, b,
                                                    (short)0, acc, false, false);
    }

    // bias + row-wise softmax (row = (reg r, half hi); columns j striped over 16 lanes)
    float bias = convb[g * 16 + lo];
    #pragma unroll
    for (int r = 0; r < 8; ++r) {
        float x = acc[r] + bias;
        float m = x;
        m = fmaxf(m, __shfl_xor(m, 1, 32));
        m = fmaxf(m, __shfl_xor(m, 2, 32));
        m = fmaxf(m, __shfl_xor(m, 4, 32));
        m = fmaxf(m, __shfl_xor(m, 8, 32));
        float e = __expf(x - m);
        float s = e;
        s += __shfl_xor(s, 1, 32);
        s += __shfl_xor(s, 2, 32);
        s += __shfl_xor(s, 4, 32);
        s += __shfl_xor(s, 8, 32);
        acc[r] = e / s;
    }

    // write softmaxed M tile: lds_M[(i*16 + g)*JP + j]
    #pragma unroll
    for (int r = 0; r < 8; ++r) {
        int i = r + hi * 8;
        lds[M_OFF + (i * 16 + g) * JP + lo] = acc[r];
    }
    __syncthreads();

    // ---- phase 2: wave w handles n = n0 + w : out = M[n] (16x16) @ R[n] (16x128) ----
    const int i = wave;
    v2f amat[4];                            // A-operand: M[n][g=lo][j0..j0+1]
    #pragma unroll
    for (int kc = 0; kc < 4; ++kc) {
        int j0 = 4 * kc + hi * 2;
        amat[kc].x = lds[M_OFF + (i * 16 + lo) * JP + j0];
        amat[kc].y = lds[M_OFF + (i * 16 + lo) * JP + j0 + 1];
    }
    float* orow = out + (n0 + i) * 2048;
    #pragma unroll
    for (int tc = 0; tc < 8; ++tc) {        // 8 column tiles of 16 along d
        v8f c = {};
        #pragma unroll
        for (int kc = 0; kc < 4; ++kc) {
            int j0 = 4 * kc + hi * 2;
            v2f b;                           // B[k][d] = R[n][j][tc*16+lo]
            b.x = lds[((j0    ) * DIN + tc * 16 + lo) * IP + i];
            b.y = lds[((j0 + 1) * DIN + tc * 16 + lo) * IP + i];
            c = __builtin_amdgcn_wmma_f32_16x16x4_f32(false, amat[kc], false, b,
                                                      (short)0, c, false, false);
        }
        #pragma unroll
        for (int r = 0; r < 8; ++r) {        // C: reg r -> row g = r + hi*8, col = lo
            orow[(r + hi * 8) * DIN + tc * 16 + lo] = c[r];
        }
    }
}

extern "C" void kernel_launch(void* const* d_in, const int* in_sizes, int n_in,
                              void* d_out, int out_size, void* d_ws, size_t ws_size,
                              hipStream_t stream) {
    const float* region = (const float*)d_in[0];
    const float* convw  = (const float*)d_in[1];
    const float* convb  = (const float*)d_in[2];
    float* out = (float*)d_out;

    int n = in_sizes[0] / (16 * DIN);        // 100000
    int blocks = n / 16;                     // 6250 tiles of 16 rows
    size_t shmem = TOTAL_LDS_FLOATS * sizeof(float);
    fused_dynconv_kernel<<<blocks, 512, shmem, stream>>>(region, convw, convb, out);
}